// FaultClassifierGNN_46368466927824
// MI455X (gfx1250) — compile-verified
//
#include <hip/hip_runtime.h>

typedef __attribute__((ext_vector_type(2))) float v2f;
typedef __attribute__((ext_vector_type(8))) float v8f;

// ---------------- utility: fill ----------------
__global__ void fill_kernel(float* __restrict__ p, long n, float val) {
  long i = (long)blockIdx.x * blockDim.x + threadIdx.x;
  if (i < n) p[i] = val;
}

// ---------------- degree (with self-loop already pre-filled as 1.0) ----------------
__global__ void degree_kernel(const int* __restrict__ dst, float* __restrict__ deg, int E) {
  int e = blockIdx.x * blockDim.x + threadIdx.x;
  if (e < E) atomicAdd(&deg[dst[e]], 1.0f);
}

// ---------------- WMMA fp32 GEMM: out[M,NDIM] = A[M,KDIM] @ W[KDIM,NDIM] ----------------
// One wave computes a 16-row x NDIM strip: each A fragment (16x4) is loaded once
// and reused across all NDIM/16 column tiles (register-resident accumulators),
// cutting A gather traffic by NDIM/16 vs one-tile-per-wave.
// Layouts per CDNA5 ISA 7.12.2 (V_WMMA_F32_16X16X4_F32):
//   A 16x4 : lanes 0-15 -> M=l, v0=K(+0), v1=K(+1); lanes 16-31 -> M=l, v0=K(+2), v1=K(+3)
//   C/D    : VGPR j, lanes 0-15 -> M=j, N=l; lanes 16-31 -> M=j+8, N=l
template <int KDIM, int NDIM>
__global__ void gemm_wmma_kernel(const float* __restrict__ A,
                                 const float* __restrict__ W,
                                 float* __restrict__ out) {
  constexpr int NT = NDIM / 16;
  const int lane = threadIdx.x & 31;
  const int wave = threadIdx.x >> 5;
  const int mt = blockIdx.x * (blockDim.x >> 5) + wave;  // row tile
  const int half = lane >> 4;  // 0 or 1: selects K pair
  const int l = lane & 15;

  const float* arow  = A + (long)(mt * 16 + l) * KDIM + 2 * half;
  const float* bbase = W + (long)(2 * half) * NDIM + l;

  v8f acc[NT];
#pragma unroll
  for (int nt = 0; nt < NT; ++nt) acc[nt] = (v8f){0.f, 0.f, 0.f, 0.f, 0.f, 0.f, 0.f, 0.f};

#pragma unroll 8
  for (int k0 = 0; k0 < KDIM; k0 += 4) {
    v2f a;
    a.x = arow[k0 + 0];
    a.y = arow[k0 + 1];
#pragma unroll
    for (int nt = 0; nt < NT; ++nt) {
      v2f b;
      b.x = bbase[(long)(k0 + 0) * NDIM + nt * 16];
      b.y = bbase[(long)(k0 + 1) * NDIM + nt * 16];
      // 8 args: (neg_a, A, neg_b, B, c_mod, C, reuse_a, reuse_b)
      acc[nt] = __builtin_amdgcn_wmma_f32_16x16x4_f32(false, a, false, b, (short)0,
                                                      acc[nt], false, false);
    }
  }

  float* obase = out + (long)(mt * 16 + 8 * half) * NDIM + l;
#pragma unroll
  for (int nt = 0; nt < NT; ++nt) {
#pragma unroll
    for (int j = 0; j < 8; ++j) obase[(long)j * NDIM + nt * 16] = acc[nt][j];
  }
}

// ---------------- edge scatter: agg[dst] += h[src] * d^-1/2 d^-1/2 ----------------
template <int HD>
__global__ void scatter_kernel(const float* __restrict__ h,
                               const float* __restrict__ deg,
                               const int* __restrict__ src,
                               const int* __restrict__ dst,
                               float* __restrict__ agg, int E) {
  long tid = (long)blockIdx.x * blockDim.x + threadIdx.x;
  long total = (long)E * HD;
  if (tid >= total) return;
  int e = (int)(tid / HD);
  int f = (int)(tid - (long)e * HD);
  int s = src[e];
  int d = dst[e];
  float norm = rsqrtf(deg[s]) * rsqrtf(deg[d]);
  atomicAdd(&agg[(long)d * HD + f], h[(long)s * HD + f] * norm);
}

// ---------------- finalize: agg = relu(agg + hpre/deg + bias) (in place) ----------------
template <int HD>
__global__ void finalize_kernel(float* __restrict__ agg,
                                const float* __restrict__ hpre,
                                const float* __restrict__ deg,
                                const float* __restrict__ bias, int Nn) {
  long tid = (long)blockIdx.x * blockDim.x + threadIdx.x;
  if (tid >= (long)Nn * HD) return;
  int row = (int)(tid / HD);
  int f = (int)(tid - (long)row * HD);
  float v = agg[tid] + hpre[tid] / deg[row] + bias[f];
  agg[tid] = v > 0.f ? v : 0.f;
}

// ---------------- global mean pool (sums + counts) ----------------
__global__ void pool_kernel(const float* __restrict__ h2, const int* __restrict__ batch,
                            float* __restrict__ sums, float* __restrict__ counts, int Nn) {
  long tid = (long)blockIdx.x * blockDim.x + threadIdx.x;
  if (tid >= (long)Nn * 64) return;
  int node = (int)(tid >> 6);
  int f = (int)(tid & 63);
  int b = batch[node];
  atomicAdd(&sums[b * 64 + f], h2[tid]);
  if (f == 0) atomicAdd(&counts[b], 1.0f);
}

// ---------------- final FC: out[50,10] = (sums/count) @ Wfc + bfc ----------------
__global__ void fc_kernel(const float* __restrict__ sums, const float* __restrict__ counts,
                          const float* __restrict__ Wfc, const float* __restrict__ bfc,
                          float* __restrict__ out) {
  int tid = threadIdx.x;
  if (tid >= 50 * 10) return;
  int b = tid / 10;
  int cc = tid % 10;
  float cnt = counts[b];
  cnt = cnt > 1.0f ? cnt : 1.0f;
  float inv = 1.0f / cnt;
  float acc = bfc[cc];
#pragma unroll 8
  for (int k = 0; k < 64; ++k) acc += sums[b * 64 + k] * inv * Wfc[k * 10 + cc];
  out[tid] = acc;
}

extern "C" void kernel_launch(void* const* d_in, const int* in_sizes, int n_in,
                              void* d_out, int out_size, void* d_ws, size_t ws_size,
                              hipStream_t stream) {
  (void)in_sizes; (void)n_in; (void)out_size; (void)ws_size;

  const float* x     = (const float*)d_in[0];
  const int*   ei    = (const int*)d_in[1];
  const int*   batch = (const int*)d_in[2];
  const float* W1    = (const float*)d_in[3];
  const float* b1    = (const float*)d_in[4];
  const float* W2    = (const float*)d_in[5];
  const float* b2    = (const float*)d_in[6];
  const float* Wfc   = (const float*)d_in[7];
  const float* bfc   = (const float*)d_in[8];
  float* out = (float*)d_out;

  const int Nn = 100000, E = 1600000, H = 128, Hh = 64, B = 50;
  const int* src = ei;
  const int* dst = ei + E;

  float* ws     = (float*)d_ws;
  float* deg    = ws;                         // Nn
  float* hpre1  = deg + Nn;                   // Nn*H
  float* agg1   = hpre1 + (long)Nn * H;       // Nn*H  (becomes h1 after finalize)
  float* hpre2  = agg1 + (long)Nn * H;        // Nn*Hh
  float* agg2   = hpre2 + (long)Nn * Hh;      // Nn*Hh (becomes h2 after finalize)
  float* sums   = agg2 + (long)Nn * Hh;       // B*Hh
  float* counts = sums + B * Hh;              // B

  // ---- init scratch (every call; atomics accumulate) ----
  fill_kernel<<<(Nn + 255) / 256, 256, 0, stream>>>(deg, Nn, 1.0f);
  fill_kernel<<<(unsigned)(((long)Nn * H + 255) / 256), 256, 0, stream>>>(agg1, (long)Nn * H, 0.0f);
  fill_kernel<<<(unsigned)(((long)Nn * Hh + 255) / 256), 256, 0, stream>>>(agg2, (long)Nn * Hh, 0.0f);
  fill_kernel<<<(B * Hh + B + 255) / 256, 256, 0, stream>>>(sums, B * Hh + B, 0.0f);

  // ---- degrees (self-loop included via 1.0 init) ----
  degree_kernel<<<(E + 255) / 256, 256, 0, stream>>>(dst, deg, E);

  // ---- layer 1: hpre1 = x @ W1 ; aggregate ; relu(agg + hpre/deg + b1) ----
  // 6250 row tiles, 2 waves per block -> 3125 blocks
  gemm_wmma_kernel<128, 128><<<(Nn / 16) / 2, 64, 0, stream>>>(x, W1, hpre1);
  scatter_kernel<128><<<(unsigned)(((long)E * 128 + 255) / 256), 256, 0, stream>>>(hpre1, deg, src, dst, agg1, E);
  finalize_kernel<128><<<(unsigned)(((long)Nn * 128 + 255) / 256), 256, 0, stream>>>(agg1, hpre1, deg, b1, Nn);

  // ---- layer 2: hpre2 = h1 @ W2 ; aggregate ; relu(agg + hpre/deg + b2) ----
  gemm_wmma_kernel<128, 64><<<(Nn / 16) / 2, 64, 0, stream>>>(agg1, W2, hpre2);
  scatter_kernel<64><<<(unsigned)(((long)E * 64 + 255) / 256), 256, 0, stream>>>(hpre2, deg, src, dst, agg2, E);
  finalize_kernel<64><<<(unsigned)(((long)Nn * 64 + 255) / 256), 256, 0, stream>>>(agg2, hpre2, deg, b2, Nn);

  // ---- mean pool + classifier head ----
  pool_kernel<<<(unsigned)(((long)Nn * 64 + 255) / 256), 256, 0, stream>>>(agg2, batch, sums, counts, Nn);
  fc_kernel<<<1, 512, 0, stream>>>(sums, counts, Wfc, bfc, out);
}